// NodeLevelEncoder_2482491097959
// MI455X (gfx1250) — compile-verified
//
#include <hip/hip_runtime.h>
#include <hip/hip_bf16.h>

#define N_NODES 50000
#define N_EDGES 800000
#define IN_CH 128
#define HID 128
#define LAT 64

typedef __attribute__((ext_vector_type(2))) float v2f;
typedef __attribute__((ext_vector_type(8))) float v8f;

// ---------------------------------------------------------------- utilities
__global__ void zero_f32_kernel(float4* __restrict__ p, int n4) {
    int i = blockIdx.x * blockDim.x + threadIdx.x;
    int stride = gridDim.x * blockDim.x;
    const float4 z = make_float4(0.f, 0.f, 0.f, 0.f);
    for (; i < n4; i += stride) p[i] = z;
}

// deg[d] += 1 for each edge (self-loop +1 folded into dinv kernel)
__global__ void degree_kernel(const long long* __restrict__ ei, float* __restrict__ deg) {
    int e = blockIdx.x * blockDim.x + threadIdx.x;
    if (e < N_EDGES) {
        int d = (int)ei[N_EDGES + e];
        atomicAdd(&deg[d], 1.0f);
    }
}

// in-place deg -> 1/sqrt(deg + 1)
__global__ void dinv_kernel(float* __restrict__ deg) {
    int i = blockIdx.x * blockDim.x + threadIdx.x;
    if (i < N_NODES) deg[i] = rsqrtf(deg[i] + 1.0f);
}

// ---------------------------------------------------------------- WMMA GEMM
// C[M x N] = A[M x 128] * B[128 x N] (+ bias). One wave computes a 16 x (16*NT)
// strip: the A fragment is loaded once per K-step and reused across NT WMMAs.
// LDB/LDC are compile-time so all B/C addressing folds to immediate offsets.
// V_WMMA_F32_16X16X4_F32: exact fp32, 32*NT WMMA steps per wave.
template <int NT, int LDB, int LDC>
__global__ __launch_bounds__(32)
void wmma_gemm_f32_kernel(const float* __restrict__ A,
                          const float* __restrict__ B,
                          const float* __restrict__ bias,  // may be null
                          float* __restrict__ C) {
    const int lane = threadIdx.x & 31;
    const int half = lane >> 4;   // 0 -> lanes 0-15 (K 0,1), 1 -> lanes 16-31 (K 2,3)
    const int l16  = lane & 15;
    const int row  = blockIdx.x * 16;          // M tile base
    const int col  = blockIdx.y * (16 * NT);   // N strip base

    const float* __restrict__ Arow = A + (long long)(row + l16) * IN_CH + half * 2;
    const float* __restrict__ Bcol = B + (long long)(half * 2) * LDB + col + l16;

    v8f acc[NT];
#pragma unroll
    for (int n = 0; n < NT; ++n) acc[n] = (v8f){};

#pragma unroll
    for (int k = 0; k < IN_CH; k += 4) {
        // A-matrix 16x4 f32 fragment (ISA 7.12.2):
        //   lanes 0-15: VGPR0=K0, VGPR1=K1 ; lanes 16-31: VGPR0=K2, VGPR1=K3
        v2f a;
        a.x = Arow[k + 0];
        a.y = Arow[k + 1];
#pragma unroll
        for (int n = 0; n < NT; ++n) {
            // B-matrix 4x16 fragment (mirrored): lane = N, VGPRs = consecutive K
            v2f b;
            b.x = Bcol[(k + 0) * LDB + n * 16];
            b.y = Bcol[(k + 1) * LDB + n * 16];
            acc[n] = __builtin_amdgcn_wmma_f32_16x16x4_f32(
                /*neg_a=*/false, a, /*neg_b=*/false, b,
                /*c_mod=*/(short)0, acc[n], /*reuse_a=*/false, /*reuse_b=*/false);
        }
    }

    float bv[NT];
#pragma unroll
    for (int n = 0; n < NT; ++n) bv[n] = bias ? bias[col + n * 16 + l16] : 0.0f;

    // C/D 16x16 f32 layout: VGPR r -> M=r (lanes 0-15) / M=r+8 (lanes 16-31)
#pragma unroll
    for (int n = 0; n < NT; ++n) {
#pragma unroll
        for (int r = 0; r < 8; ++r) {
            C[(long long)(row + r + half * 8) * LDC + col + n * 16 + l16] =
                acc[n][r] + bv[n];
        }
    }
}

// ---------------------------------------------------------------- scatter
// One wave per edge; each lane handles 4 channels (b128 gather + 4 f32 atomics).
// agg[dst,c] += t[src,c] * dinv[src]*dinv[dst]   (global_atomic_add_f32, L2)
__global__ __launch_bounds__(256)
void scatter_kernel(const long long* __restrict__ ei,
                    const float* __restrict__ t,
                    const float* __restrict__ dinv,
                    float* __restrict__ agg) {
    const int tid = blockIdx.x * 256 + threadIdx.x;
    const int e = tid >> 5;                 // edge (one wave per edge)
    const int c = (tid & 31) * 4;           // channel base for this lane
    if (e < N_EDGES) {
        const int s = (int)ei[e];
        const int d = (int)ei[N_EDGES + e];
        const float norm = dinv[s] * dinv[d];
        const float4 v = *(const float4*)(t + (long long)s * HID + c);
        float* dst = agg + (long long)d * HID + c;
        atomicAdd(dst + 0, v.x * norm);
        atomicAdd(dst + 1, v.y * norm);
        atomicAdd(dst + 2, v.z * norm);
        atomicAdd(dst + 3, v.w * norm);
    }
}

// ---------------------------------------------------------------- finalize
// h = relu(agg + t * dinv^2 + b), float4-vectorized
__global__ void finalize_kernel(const float4* __restrict__ agg,
                                const float4* __restrict__ t,
                                const float* __restrict__ dinv,
                                const float* __restrict__ bias,
                                float4* __restrict__ h) {
    const int i = blockIdx.x * blockDim.x + threadIdx.x;   // float4 index
    const int total4 = N_NODES * (HID / 4);                // 1.6M
    if (i < total4) {
        const int node = i >> 5;            // 32 float4 per node
        const int c    = (i & 31) * 4;
        float sn = dinv[node];
        sn *= sn;
        const float4 av = agg[i];
        const float4 tv = t[i];
        float4 r;
        r.x = av.x + tv.x * sn + bias[c + 0];
        r.y = av.y + tv.y * sn + bias[c + 1];
        r.z = av.z + tv.z * sn + bias[c + 2];
        r.w = av.w + tv.w * sn + bias[c + 3];
        r.x = r.x > 0.f ? r.x : 0.f;
        r.y = r.y > 0.f ? r.y : 0.f;
        r.z = r.z > 0.f ? r.z : 0.f;
        r.w = r.w > 0.f ? r.w : 0.f;
        h[i] = r;
    }
}

// ---------------------------------------------------------------- launcher
extern "C" void kernel_launch(void* const* d_in, const int* in_sizes, int n_in,
                              void* d_out, int out_size, void* d_ws, size_t ws_size,
                              hipStream_t stream) {
    const float*     x    = (const float*)d_in[0];
    const long long* ei   = (const long long*)d_in[1];   // int64 [2, E]
    const float*     W1   = (const float*)d_in[2];
    const float*     b1   = (const float*)d_in[3];
    const float*     W2   = (const float*)d_in[4];
    const float*     b2   = (const float*)d_in[5];
    const float*     Wmu  = (const float*)d_in[6];
    const float*     bmu  = (const float*)d_in[7];
    const float*     Wlv  = (const float*)d_in[8];
    const float*     blv  = (const float*)d_in[9];
    float* out = (float*)d_out;

    // workspace layout (floats): dinv | t | agg | h   (~77 MB)
    float* dinv = (float*)d_ws;
    float* t    = dinv + 50048;                       // 16B-aligned pad
    float* agg  = t    + (long long)N_NODES * HID;
    float* h    = agg  + (long long)N_NODES * HID;

    const int NH  = N_NODES * HID;                    // 6.4M floats
    const int MT  = N_NODES / 16;                     // 3125 M-tiles (exact)
    const int SCB = (N_EDGES * 32 + 255) / 256;       // scatter blocks (wave/edge)
    const int FNB = (NH / 4 + 255) / 256;             // finalize blocks

    // --- gcn_norm: deg -> dinv (reuse buffer in place) ---
    zero_f32_kernel<<<64, 256, 0, stream>>>((float4*)dinv, 50048 / 4);
    degree_kernel<<<(N_EDGES + 255) / 256, 256, 0, stream>>>(ei, dinv);
    dinv_kernel<<<(N_NODES + 255) / 256, 256, 0, stream>>>(dinv);

    // --- layer 1: t = x @ W1 ; scatter ; h = relu(agg + t*dinv^2 + b1)
    wmma_gemm_f32_kernel<4, HID, HID><<<dim3(MT, 2), 32, 0, stream>>>(x, W1, nullptr, t);
    zero_f32_kernel<<<2048, 256, 0, stream>>>((float4*)agg, NH / 4);
    scatter_kernel<<<SCB, 256, 0, stream>>>(ei, t, dinv, agg);
    finalize_kernel<<<FNB, 256, 0, stream>>>((const float4*)agg, (const float4*)t,
                                             dinv, b1, (float4*)h);

    // --- layer 2: t = h @ W2 ; scatter ; h = relu(agg + t*dinv^2 + b2)
    wmma_gemm_f32_kernel<4, HID, HID><<<dim3(MT, 2), 32, 0, stream>>>(h, W2, nullptr, t);
    zero_f32_kernel<<<2048, 256, 0, stream>>>((float4*)agg, NH / 4);
    scatter_kernel<<<SCB, 256, 0, stream>>>(ei, t, dinv, agg);
    finalize_kernel<<<FNB, 256, 0, stream>>>((const float4*)agg, (const float4*)t,
                                             dinv, b2, (float4*)h);

    // --- heads: mu = h @ W_mu + b_mu ; logvar = h @ W_lv + b_lv (into d_out)
    wmma_gemm_f32_kernel<4, LAT, LAT><<<dim3(MT, 1), 32, 0, stream>>>(h, Wmu, bmu, out);
    wmma_gemm_f32_kernel<4, LAT, LAT><<<dim3(MT, 1), 32, 0, stream>>>(
        h, Wlv, blv, out + (long long)N_NODES * LAT);
}